// CostVolume_82008105550452
// MI455X (gfx1250) — compile-verified
//
#include <hip/hip_runtime.h>
#include <hip/hip_bf16.h>

typedef __attribute__((ext_vector_type(2))) float v2f;
typedef __attribute__((ext_vector_type(8))) float v8f;

namespace {
constexpr int Bn = 8;
constexpr int Cn = 256;
constexpr int Hn = 96;
constexpr int Wn = 320;
constexpr int Sn = 9;            // 2*D+1 shifts
constexpr int HW = Hn * Wn;      // 30720
constexpr long CHW = (long)Cn * HW;
constexpr int WTILES = Wn / 16;  // 20
constexpr int WAVES_PER_BLOCK = 8;
constexpr int LDS_STRIDE = 33;   // 16x32 tile padded to avoid bank conflicts
}

__global__ __launch_bounds__(256)
void CostVolume_wmma_kernel(const float* __restrict__ f1,
                            const float* __restrict__ f2,
                            float* __restrict__ out) {
    __shared__ float lds[WAVES_PER_BLOCK][16 * LDS_STRIDE];

    const int lane = threadIdx.x & 31;
    const int wv   = threadIdx.x >> 5;
    const int tile = blockIdx.x * WAVES_PER_BLOCK + wv;   // 0 .. 15359

    const int wt  = tile % WTILES;
    const int rem = tile / WTILES;
    const int h   = rem % Hn;
    const int b   = rem / Hn;
    const int w0  = wt * 16;

    const int lhalf = lane >> 4;     // 0 | 1  -> K rows {0,1} | {2,3}
    const int ln    = lane & 15;     // M row (A) / N col (B)
    const int kr0   = lhalf * 2;

    // A operand: f1[b, kr0+{0,1} + c0, h, w0+ln]   (always in range)
    const float* p1 = f1 + (long)b * CHW + (long)kr0 * HW + (long)h * Wn + (w0 + ln);

    // B operands: two 16-wide column tiles of f2, zero-padded outside [0,W).
    // Clamp the address so it is always legal; zero-padding is applied with a
    // branch-free select (loop-invariant per-lane mask -> v_cndmask, no exec
    // branch diamonds in the inner loop).
    const int col1 = w0 - 4 + ln;        // N-tile 0: cols w0-4 .. w0+11
    const int col2 = w0 + 12 + ln;       // N-tile 1: cols w0+12 .. w0+27
    const bool ok1 = (col1 >= 0) && (col1 < Wn);
    const bool ok2 = (col2 < Wn);
    const int col1c = ok1 ? col1 : 0;
    const int col2c = ok2 ? col2 : 0;
    const float* p2a = f2 + (long)b * CHW + (long)kr0 * HW + (long)h * Wn + col1c;
    const float* p2b = f2 + (long)b * CHW + (long)kr0 * HW + (long)h * Wn + col2c;

    v8f acc1 = {};
    v8f acc2 = {};

#pragma unroll 4
    for (int c0 = 0; c0 < Cn; c0 += 4) {
        v2f a;
        a.x = p1[0];
        a.y = p1[HW];

        const float b1x = p2a[0];
        const float b1y = p2a[HW];
        const float b2x = p2b[0];
        const float b2y = p2b[HW];

        v2f b1;
        b1.x = ok1 ? b1x : 0.0f;
        b1.y = ok1 ? b1y : 0.0f;

        v2f b2;
        b2.x = ok2 ? b2x : 0.0f;
        b2.y = ok2 ? b2y : 0.0f;

        // D = A(16x4) * B(4x16) + C, fp32 in / fp32 accum
        acc1 = __builtin_amdgcn_wmma_f32_16x16x4_f32(
            /*neg_a=*/false, a, /*neg_b=*/false, b1,
            /*c_mod=*/(short)0, acc1, /*reuse_a=*/false, /*reuse_b=*/false);
        acc2 = __builtin_amdgcn_wmma_f32_16x16x4_f32(
            /*neg_a=*/false, a, /*neg_b=*/false, b2,
            /*c_mod=*/(short)0, acc2, /*reuse_a=*/false, /*reuse_b=*/false);

        p1  += 4 * HW;
        p2a += 4 * HW;
        p2b += 4 * HW;
    }

    // Stage 16x32 fp32 result tile to LDS.
    // C/D layout: VGPR v, lanes 0-15 -> (M=v,   N=ln), lanes 16-31 -> (M=v+8, N=ln)
    float* t = lds[wv];
#pragma unroll
    for (int v = 0; v < 8; ++v) {
        const int m = v + 8 * lhalf;
        t[m * LDS_STRIDE + ln]      = acc1[v];
        t[m * LDS_STRIDE + 16 + ln] = acc2[v];
    }
    __syncthreads();   // each wave reads only its own tile, but keep ordering simple/safe

    // out[b, i, h, w0+m] = tile[m][m + i] / C
    const float scale = 1.0f / (float)Cn;
    if (lane < 16) {
        const int m = lane;
        float* op = out + (long)b * Sn * HW + (long)h * Wn + (w0 + m);
#pragma unroll
        for (int i = 0; i < Sn; ++i) {
            op[(long)i * HW] = t[m * LDS_STRIDE + m + i] * scale;
        }
    }
}

extern "C" void kernel_launch(void* const* d_in, const int* in_sizes, int n_in,
                              void* d_out, int out_size, void* d_ws, size_t ws_size,
                              hipStream_t stream) {
    (void)in_sizes; (void)n_in; (void)out_size; (void)d_ws; (void)ws_size;
    const float* f1 = (const float*)d_in[0];
    const float* f2 = (const float*)d_in[1];
    float* out = (float*)d_out;

    const int total_tiles = Bn * Hn * WTILES;              // 15360 waves
    const int blocks = total_tiles / WAVES_PER_BLOCK;      // 1920 blocks of 256 threads
    CostVolume_wmma_kernel<<<blocks, 256, 0, stream>>>(f1, f2, out);
}